// AMP_net_Deblock_73589969649846
// MI455X (gfx1250) — compile-verified
//
#include <hip/hip_runtime.h>
#include <hip/hip_bf16.h>

// ---------------------------------------------------------------------------
// AMP-Net deblock pipeline for gfx1250 (MI455X).
// All dense math runs through V_WMMA_F32_16X16X4_F32 (native fp32 tensor op,
// preserves fp32 reference numerics). 32->32 convs are implicit GEMM.
// GEMM-B and conv LDS layouts are K-pair interleaved so each such fragment is a single
// aligned ds_load_b64 into an even VGPR pair (no repack movs).
// ---------------------------------------------------------------------------

typedef float v2f __attribute__((ext_vector_type(2)));
typedef float v8f __attribute__((ext_vector_type(8)));

__device__ __forceinline__ v8f wmma_f32(v2f a, v2f b, v8f c) {
  // D = A(16x4) * B(4x16) + C(16x16), fp32.
  return __builtin_amdgcn_wmma_f32_16x16x4_f32(false, a, false, b,
                                               (short)0, c, false, false);
}

__device__ __forceinline__ v2f lds_pair(const float* p) {
  return *(const v2f*)p;   // 8-byte aligned except GEMM As rows (stride 33 floats)
}

#define BQ   33
#define BB   1089      // 33*33
#define NCOL 512       // L*H*S = 8*8*8
#define MROW 272
#define SIMG 8
#define WIMG 264       // 8*33

// ---------------------------------------------------------------------------
// blockify: img [S][264][264] -> X [1089][512], col n = l*64 + h*8 + s
// ---------------------------------------------------------------------------
__global__ void k_blockify(const float* __restrict__ img, float* __restrict__ X) {
  int idx = blockIdx.x * blockDim.x + threadIdx.x;
  if (idx >= BB * NCOL) return;
  int n = idx % NCOL;
  int r = idx / NCOL;
  int s = n & 7, h = (n >> 3) & 7, l = n >> 6;
  int bi = r / BQ, bj = r % BQ;
  X[idx] = img[(size_t)s * WIMG * WIMG + (h * BQ + bi) * WIMG + (l * BQ + bj)];
}

__global__ void k_unblockify(const float* __restrict__ X, float* __restrict__ img) {
  int idx = blockIdx.x * blockDim.x + threadIdx.x;
  if (idx >= SIMG * WIMG * WIMG) return;
  int xx = idx % WIMG;
  int yy = (idx / WIMG) % WIMG;
  int s  = idx / (WIMG * WIMG);
  int h = yy / BQ, bi = yy % BQ;
  int l = xx / BQ, bj = xx % BQ;
  int n = l * 64 + h * 8 + s;
  int r = bi * BQ + bj;
  img[idx] = X[(size_t)r * NCOL + n];
}

// ---------------------------------------------------------------------------
// LDS-tiled transpose: dst[c][r] = src[r][c]
// ---------------------------------------------------------------------------
__global__ __launch_bounds__(128)
void k_transpose(const float* __restrict__ src, float* __restrict__ dst,
                 int rows, int cols) {
  __shared__ float tile[32][33];
  int tx = threadIdx.x & 31, ty = threadIdx.x >> 5;
  int bx = blockIdx.x * 32, by = blockIdx.y * 32;
  for (int i = ty; i < 32; i += 4) {
    int r = by + i, c = bx + tx;
    tile[i][tx] = (r < rows && c < cols) ? src[(size_t)r * cols + c] : 0.f;
  }
  __syncthreads();
  for (int i = ty; i < 32; i += 4) {
    int r = bx + i, c = by + tx;     // dst is [cols][rows]
    if (r < cols && c < rows) dst[(size_t)r * rows + c] = tile[tx][i];
  }
}

// ---------------------------------------------------------------------------
// WMMA fp32 GEMM: C = epilogue(A[MxK] @ B[KxN])
//   mode 0: C = acc
//   mode 1: C = D0 - acc                     (r = y - A@X)
//   mode 2: C = s*acc + D0                   (z = s*At r + X)
//   mode 3: C = D0 - s*acc + D1              (X = z - s*AtA@noise + noise)
// s = steps[layer] read on-device (graph-capture safe).
// Block: 128 threads = 4 waves; tile 64x64; each wave: 16 rows x 64 cols.
// B tile is K-pair interleaved: Bs[kp][col][j] = B[2*kp+j][col].
// ---------------------------------------------------------------------------
#define GBM 64
#define GBN 64
#define GBK 32

__global__ __launch_bounds__(128)
void k_gemm_wmma(const float* __restrict__ Am, const float* __restrict__ Bm,
                 const float* __restrict__ D0, const float* __restrict__ D1,
                 float* __restrict__ C, int M, int N, int K,
                 int mode, const float* __restrict__ steps, int layer) {
  __shared__ float As[GBM][GBK + 1];        // row-major, K contiguous
  __shared__ float Bs[GBK / 2][GBN][2];     // K-pair interleaved
  const int tid  = threadIdx.x;
  const int wave = tid >> 5;
  const int lane = tid & 31;
  const int bn = blockIdx.x * GBN;
  const int bm = blockIdx.y * GBM;

  v8f acc[4];
  #pragma unroll
  for (int i = 0; i < 4; ++i) acc[i] = (v8f){0.f,0.f,0.f,0.f,0.f,0.f,0.f,0.f};

  const int mrow = wave * 16 + (lane & 15);
  const int ks   = (lane >> 4) * 2;   // 0 or 2
  const int hi   = lane >> 4;         // 0 or 1
  const int colL = lane & 15;

  for (int k0 = 0; k0 < K; k0 += GBK) {
    // stage A tile 64x32 (K contiguous per row)
    for (int i = tid; i < GBM * GBK; i += 128) {
      int r = i >> 5, c = i & 31;
      int gr = bm + r, gc = k0 + c;
      As[r][c] = (gr < M && gc < K) ? Am[(size_t)gr * K + gc] : 0.f;
    }
    // stage B tile 32x64, interleaved by K pairs
    for (int i = tid; i < GBK * GBN; i += 128) {
      int r = i >> 6, c = i & 63;
      int gr = k0 + r, gc = bn + c;
      Bs[r >> 1][c][r & 1] = (gr < K && gc < N) ? Bm[(size_t)gr * N + gc] : 0.f;
    }
    // hint next A tile into cache
    if (k0 + GBK < K && bm + (tid >> 1) < M)
      __builtin_prefetch(&Am[(size_t)(bm + (tid >> 1)) * K + k0 + GBK], 0, 0);
    __syncthreads();

    #pragma unroll
    for (int kk = 0; kk < GBK; kk += 4) {
      v2f a = lds_pair(&As[mrow][kk + ks]);
      const int kp = (kk >> 1) + hi;            // (kk+ks)/2
      #pragma unroll
      for (int nt = 0; nt < 4; ++nt) {
        v2f b = lds_pair(&Bs[kp][nt * 16 + colL][0]);
        acc[nt] = wmma_f32(a, b, acc[nt]);
      }
    }
    __syncthreads();
  }

  float s = 1.f;
  if (steps) s = steps[layer];
  #pragma unroll
  for (int nt = 0; nt < 4; ++nt) {
    int gcol = bn + nt * 16 + colL;
    if (gcol >= N) continue;
    #pragma unroll
    for (int i = 0; i < 8; ++i) {
      int grow = bm + wave * 16 + i + 8 * hi;
      if (grow >= M) continue;
      size_t off = (size_t)grow * N + gcol;
      float v = acc[nt][i];
      float o;
      if      (mode == 1) o = D0[off] - v;
      else if (mode == 2) o = s * v + D0[off];
      else if (mode == 3) o = D0[off] - s * v + D1[off];
      else                o = v;
      C[off] = o;
    }
  }
}

// ---------------------------------------------------------------------------
// 3x3 SAME conv, Cin=Cout=32, implicit GEMM on WMMA f32 16x16x4.
// Per block: one image, 8-row x 16-col output tile, all 32 out channels.
// 4 waves; each wave: 2 output rows x 2 oc-tiles = 4 accumulators.
// K loop = 9 taps x 32 in-channels (step 4) -> 288 WMMA per wave.
// Both LDS operands are ic-pair interleaved -> single b64 fragment loads.
// ---------------------------------------------------------------------------
__global__ __launch_bounds__(128)
void k_conv3x3_wmma(const float* __restrict__ in, const float* __restrict__ w,
                    const float* __restrict__ bias, float* __restrict__ out,
                    int Nimg, int H, int W, int relu) {
  __shared__ float patch2[16][10][20][2];   // [icp][py][px][pair]
  __shared__ float wl2[9][16][33][2];       // [tap][icp][oc pad33][pair]

  const int tid  = threadIdx.x;
  const int wave = tid >> 5;
  const int lane = tid & 31;
  const int tilesX = (W + 15) / 16;
  const int tilesY = (H + 7) / 8;
  int b   = blockIdx.x;
  int img = b / (tilesX * tilesY);
  int t   = b % (tilesX * tilesY);
  int ty0 = (t / tilesX) * 8;
  int tx0 = (t % tilesX) * 16;
  const size_t plane = (size_t)H * W;
  const float* inI = in + (size_t)img * 32 * plane;

  // stage weights: [oc][ic][tap] -> [tap][ic/2][oc][ic&1]
  for (int i = tid; i < 9 * 32 * 32; i += 128) {
    int oc = i & 31, ic = (i >> 5) & 31, tap = i >> 10;
    wl2[tap][ic >> 1][oc][ic & 1] = w[(size_t)(oc * 32 + ic) * 9 + tap];
  }
  // stage input patch with zero halo: 32 ch x 10 rows x 18 cols
  for (int i = tid; i < 32 * 10 * 18; i += 128) {
    int px = i % 18, py = (i / 18) % 10, ic = i / 180;
    int gy = ty0 + py - 1, gx = tx0 + px - 1;
    float v = 0.f;
    if (gy >= 0 && gy < H && gx >= 0 && gx < W)
      v = inI[(size_t)ic * plane + (size_t)gy * W + gx];
    patch2[ic >> 1][py][px][ic & 1] = v;
  }
  __syncthreads();

  v8f acc[2][2];
  #pragma unroll
  for (int i = 0; i < 2; ++i)
    #pragma unroll
    for (int j = 0; j < 2; ++j)
      acc[i][j] = (v8f){0.f,0.f,0.f,0.f,0.f,0.f,0.f,0.f};

  const int r0   = wave * 2;          // this wave's two output rows in tile
  const int colL = lane & 15;
  const int hi   = lane >> 4;         // selects K pair within the 4-wide step

  #pragma unroll
  for (int tap = 0; tap < 9; ++tap) {
    const int dy = tap / 3, dx = tap % 3;
    #pragma unroll
    for (int k = 0; k < 32; k += 4) {
      const int kp = (k >> 1) + hi;   // (k+ks)/2
      v2f a0 = lds_pair(&wl2[tap][kp][colL][0]);
      v2f a1 = lds_pair(&wl2[tap][kp][16 + colL][0]);
      v2f b0 = lds_pair(&patch2[kp][r0 + dy][dx + colL][0]);
      v2f b1 = lds_pair(&patch2[kp][r0 + 1 + dy][dx + colL][0]);
      acc[0][0] = wmma_f32(a0, b0, acc[0][0]);
      acc[0][1] = wmma_f32(a1, b0, acc[0][1]);
      acc[1][0] = wmma_f32(a0, b1, acc[1][0]);
      acc[1][1] = wmma_f32(a1, b1, acc[1][1]);
    }
  }

  const int gx = tx0 + colL;
  float* outI = out + (size_t)img * 32 * plane;
  #pragma unroll
  for (int ri = 0; ri < 2; ++ri) {
    int gy = ty0 + r0 + ri;
    if (gy >= H || gx >= W) continue;
    #pragma unroll
    for (int ot = 0; ot < 2; ++ot) {
      #pragma unroll
      for (int i = 0; i < 8; ++i) {
        int oc = ot * 16 + i + 8 * hi;
        float v = acc[ri][ot][i] + bias[oc];
        if (relu) v = fmaxf(v, 0.f);
        outI[(size_t)oc * plane + (size_t)gy * W + gx] = v;
      }
    }
  }
}

// ---------------------------------------------------------------------------
// conv 1 -> 32 (bias + relu). One thread per pixel, loops out channels.
// ---------------------------------------------------------------------------
__global__ void k_conv3x3_in1(const float* __restrict__ in,
                              const float* __restrict__ w,
                              const float* __restrict__ bias,
                              float* __restrict__ out,
                              int Nimg, int H, int W) {
  int idx = blockIdx.x * blockDim.x + threadIdx.x;
  int tot = Nimg * H * W;
  if (idx >= tot) return;
  int x = idx % W, y = (idx / W) % H, img = idx / (W * H);
  const size_t plane = (size_t)H * W;
  const float* inI = in + (size_t)img * plane;
  float v[9];
  #pragma unroll
  for (int t = 0; t < 9; ++t) {
    int yy = y + t / 3 - 1, xx = x + t % 3 - 1;
    v[t] = (yy >= 0 && yy < H && xx >= 0 && xx < W) ? inI[(size_t)yy * W + xx] : 0.f;
  }
  float* outI = out + (size_t)img * 32 * plane + (size_t)y * W + x;
  for (int oc = 0; oc < 32; ++oc) {
    float s = bias[oc];
    #pragma unroll
    for (int t = 0; t < 9; ++t) s = fmaf(w[oc * 9 + t], v[t], s);
    outI[(size_t)oc * plane] = fmaxf(s, 0.f);
  }
}

// ---------------------------------------------------------------------------
// conv 32 -> 1, no bias. If base != null: out = base - conv (deblock residual).
// ---------------------------------------------------------------------------
__global__ void k_conv3x3_out1(const float* __restrict__ in,
                               const float* __restrict__ w,
                               const float* __restrict__ base,
                               float* __restrict__ out,
                               int Nimg, int H, int W) {
  int idx = blockIdx.x * blockDim.x + threadIdx.x;
  int tot = Nimg * H * W;
  if (idx >= tot) return;
  int x = idx % W, y = (idx / W) % H, img = idx / (W * H);
  const size_t plane = (size_t)H * W;
  const float* inI = in + (size_t)img * 32 * plane;
  float s = 0.f;
  for (int ic = 0; ic < 32; ++ic) {
    const float* p  = inI + (size_t)ic * plane;
    const float* wk = w + ic * 9;
    #pragma unroll
    for (int t = 0; t < 9; ++t) {
      int yy = y + t / 3 - 1, xx = x + t % 3 - 1;
      if (yy >= 0 && yy < H && xx >= 0 && xx < W)
        s = fmaf(wk[t], p[(size_t)yy * W + xx], s);
    }
  }
  out[idx] = base ? (base[idx] - s) : s;
}

// ---------------------------------------------------------------------------
// Host: orchestrate the whole forward pass on `stream`.
// ---------------------------------------------------------------------------
extern "C" void kernel_launch(void* const* d_in, const int* in_sizes, int n_in,
                              void* d_out, int out_size, void* d_ws, size_t ws_size,
                              hipStream_t stream) {
  const float* inputs = (const float*)d_in[0];   // [8,1,264,264]
  const float* A      = (const float*)d_in[1];   // [272,1089]
  const float* Q      = (const float*)d_in[2];   // [1089,272]
  const float* steps  = (const float*)d_in[3];   // [layers]
  const float* den_w1 = (const float*)d_in[4];
  const float* den_b1 = (const float*)d_in[5];
  const float* den_w2 = (const float*)d_in[6];
  const float* den_b2 = (const float*)d_in[7];
  const float* den_w3 = (const float*)d_in[8];
  const float* den_b3 = (const float*)d_in[9];
  const float* den_w4 = (const float*)d_in[10];
  const float* deb_w1 = (const float*)d_in[11];
  const float* deb_b1 = (const float*)d_in[12];
  const float* deb_w2 = (const float*)d_in[13];
  const float* deb_b2 = (const float*)d_in[14];
  const float* deb_w3 = (const float*)d_in[15];
  const float* deb_b3 = (const float*)d_in[16];
  const float* deb_w4 = (const float*)d_in[17];
  const int layers = in_sizes[3];

  const int PIX = BB * NCOL;                 // 557568 total pixels / X elements

  // carve workspace (floats)
  float* ws = (float*)d_ws;
  size_t o = 0;
  float* Xb     = ws + o; o += (size_t)BB * NCOL;
  float* yv     = ws + o; o += (size_t)MROW * NCOL;
  float* Xm     = ws + o; o += (size_t)BB * NCOL;
  float* AtA    = ws + o; o += (size_t)BB * BB;
  float* rv     = ws + o; o += (size_t)MROW * NCOL;
  float* zv     = ws + o; o += (size_t)BB * NCOL;
  float* noiseX = ws + o; o += (size_t)BB * NCOL;
  float* imgP   = ws + o; o += (size_t)PIX;               // image-layout buf
  float* imgQ   = ws + o; o += (size_t)PIX;               // image-layout buf
  float* actA   = ws + o; o += (size_t)PIX * 32;
  float* actB   = ws + o; o += (size_t)PIX * 32;
  (void)ws_size; (void)n_in; (void)out_size;

  const int ELT = 256;
  auto gemm = [&](const float* a, const float* b, const float* d0,
                  const float* d1, float* c, int M, int N, int K,
                  int mode, const float* st, int layer) {
    dim3 grid((N + GBN - 1) / GBN, (M + GBM - 1) / GBM);
    k_gemm_wmma<<<grid, 128, 0, stream>>>(a, b, d0, d1, c, M, N, K, mode, st, layer);
  };

  // ---- setup -------------------------------------------------------------
  k_blockify<<<(PIX + ELT - 1) / ELT, ELT, 0, stream>>>(inputs, Xb);
  gemm(A, Xb, nullptr, nullptr, yv, MROW, NCOL, BB, 0, nullptr, 0);   // y = A @ Xb
  gemm(Q, yv, nullptr, nullptr, Xm, BB, NCOL, MROW, 0, nullptr, 0);   // X = Q @ y
  gemm(Q, A,  nullptr, nullptr, AtA, BB, BB, MROW, 0, nullptr, 0);    // AtA = At@A

  for (int n = 0; n < layers; ++n) {
    // r = y - A@X ; z = s*(At@r) + X
    gemm(A, Xm, yv, nullptr, rv, MROW, NCOL, BB, 1, nullptr, 0);
    gemm(Q, rv, Xm, nullptr, zv, BB, NCOL, MROW, 2, steps, n);

    // ---- denoiser CNN on X (blocks as 512 images of 33x33) --------------
    {
      dim3 tg((NCOL + 31) / 32, (BB + 31) / 32);
      k_transpose<<<tg, 128, 0, stream>>>(Xm, imgP, BB, NCOL);  // [512][1089]
    }
    const int dH = BQ, dW = BQ, dN = NCOL;
    int dTiles = dN * ((dW + 15) / 16) * ((dH + 7) / 8);
    k_conv3x3_in1<<<(PIX + ELT - 1) / ELT, ELT, 0, stream>>>(
        imgP, den_w1 + (size_t)n * 32 * 9, den_b1 + n * 32, actA, dN, dH, dW);
    k_conv3x3_wmma<<<dTiles, 128, 0, stream>>>(
        actA, den_w2 + (size_t)n * 32 * 32 * 9, den_b2 + n * 32, actB, dN, dH, dW, 1);
    k_conv3x3_wmma<<<dTiles, 128, 0, stream>>>(
        actB, den_w3 + (size_t)n * 32 * 32 * 9, den_b3 + n * 32, actA, dN, dH, dW, 1);
    k_conv3x3_out1<<<(PIX + ELT - 1) / ELT, ELT, 0, stream>>>(
        actA, den_w4 + (size_t)n * 32 * 9, nullptr, imgQ, dN, dH, dW);
    {
      dim3 tg((BB + 31) / 32, (NCOL + 31) / 32);
      k_transpose<<<tg, 128, 0, stream>>>(imgQ, noiseX, NCOL, BB); // [1089][512]
    }

    // X = z - s*(AtA @ noise) + noise
    gemm(AtA, noiseX, zv, noiseX, Xm, BB, NCOL, BB, 3, steps, n);

    // ---- deblocker CNN on full images (8 x 264 x 264) -------------------
    k_unblockify<<<(PIX + ELT - 1) / ELT, ELT, 0, stream>>>(Xm, imgP);
    const int bH = WIMG, bW = WIMG, bN = SIMG;
    int bTiles = bN * ((bW + 15) / 16) * ((bH + 7) / 8);
    k_conv3x3_in1<<<(PIX + ELT - 1) / ELT, ELT, 0, stream>>>(
        imgP, deb_w1 + (size_t)n * 32 * 9, deb_b1 + n * 32, actA, bN, bH, bW);
    k_conv3x3_wmma<<<bTiles, 128, 0, stream>>>(
        actA, deb_w2 + (size_t)n * 32 * 32 * 9, deb_b2 + n * 32, actB, bN, bH, bW, 1);
    k_conv3x3_wmma<<<bTiles, 128, 0, stream>>>(
        actB, deb_w3 + (size_t)n * 32 * 32 * 9, deb_b3 + n * 32, actA, bN, bH, bW, 1);
    // img = img - deblock(img)  (residual fused into final conv)
    k_conv3x3_out1<<<(PIX + ELT - 1) / ELT, ELT, 0, stream>>>(
        actA, deb_w4 + (size_t)n * 32 * 9, imgP, imgQ, bN, bH, bW);
    k_blockify<<<(PIX + ELT - 1) / ELT, ELT, 0, stream>>>(imgQ, Xm);
  }

  k_unblockify<<<(PIX + ELT - 1) / ELT, ELT, 0, stream>>>(Xm, (float*)d_out);
}